// CTCPerSpeakerExtractorConcatSoftmax_63462436765738
// MI455X (gfx1250) — compile-verified
//
#include <hip/hip_runtime.h>
#include <hip/hip_bf16.h>
#include <math.h>

typedef float v2f __attribute__((ext_vector_type(2)));
typedef float v8f __attribute__((ext_vector_type(8)));

__device__ __forceinline__ float gelu_exact(float x) {
  return 0.5f * x * (1.0f + erff(x * 0.70710678118654752f));
}

#define WMMA_F32(cacc, av, bv)                                                 \
  cacc = __builtin_amdgcn_wmma_f32_16x16x4_f32(false, (av), false, (bv),       \
                                               (short)0, (cacc), false, false)

// ---------------------------------------------------------------------------
// fp32 WMMA GEMM:  C[M,N] = A[M,Kd] @ W[N,Kd]^T + bias (+ epilogue)
// flags: bit0 = add residual R[M,N]; bit1 = exact GELU; bit2 = speaker-concat
//        output row mapping out_row = b*Kspk*T + kk*T + (row%T)
// Wave tile 16x64 (4 v_wmma accumulator chains for ILP), block = 8 waves
// arranged 4x2 -> 64x128 block tile. Inner loop is 2-stage software
// pipelined so fragment loads overlap the wmma groups.
// Requires: M % 64 == 0, N % 128 == 0, Kd % 8 == 0 (all true here).
// ---------------------------------------------------------------------------
__global__ __launch_bounds__(256) void gemm_wmma_f32(
    const float* __restrict__ A, const float* __restrict__ W,
    const float* __restrict__ bias, const float* __restrict__ R,
    float* __restrict__ C, int M, int N, int Kd,
    int flags, int Tlen, int Kspk, int kk)
{
  const int lane = threadIdx.x & 31;
  const int w    = threadIdx.x >> 5;
  const int wm   = w & 3;            // 4 wave-rows
  const int wn   = w >> 2;           // 2 wave-cols
  const int tileM = blockIdx.x * 64 + wm * 16;
  const int tileN = blockIdx.y * 128 + wn * 64;
  const int half = lane >> 4;        // k-half select
  const int l16  = lane & 15;

  const int mrow = tileM + l16;
  const int n0   = tileN + l16;

  // A fragment: lane holds A[m][k0+2h], A[m][k0+2h+1]
  // B fragment: lane holds W[n][k0+2h], W[n][k0+2h+1]
  const float* Ap = A + (size_t)mrow * Kd + 2 * half;
  const float* Bp0 = W + (size_t)(n0)      * Kd + 2 * half;
  const float* Bp1 = W + (size_t)(n0 + 16) * Kd + 2 * half;
  const float* Bp2 = W + (size_t)(n0 + 32) * Kd + 2 * half;
  const float* Bp3 = W + (size_t)(n0 + 48) * Kd + 2 * half;

  v8f c0 = {}, c1 = {}, c2 = {}, c3 = {};

  // 2-stage software pipeline: stage0 holds data(k), stage1 holds data(k+4)
  v2f a0  = *(const v2f*)(Ap);
  v2f b00 = *(const v2f*)(Bp0);
  v2f b01 = *(const v2f*)(Bp1);
  v2f b02 = *(const v2f*)(Bp2);
  v2f b03 = *(const v2f*)(Bp3);
  v2f a1  = *(const v2f*)(Ap + 4);
  v2f b10 = *(const v2f*)(Bp0 + 4);
  v2f b11 = *(const v2f*)(Bp1 + 4);
  v2f b12 = *(const v2f*)(Bp2 + 4);
  v2f b13 = *(const v2f*)(Bp3 + 4);

  for (int k = 0; k < Kd - 8; k += 8) {
    WMMA_F32(c0, a0, b00);
    WMMA_F32(c1, a0, b01);
    WMMA_F32(c2, a0, b02);
    WMMA_F32(c3, a0, b03);
    a0  = *(const v2f*)(Ap  + k + 8);
    b00 = *(const v2f*)(Bp0 + k + 8);
    b01 = *(const v2f*)(Bp1 + k + 8);
    b02 = *(const v2f*)(Bp2 + k + 8);
    b03 = *(const v2f*)(Bp3 + k + 8);
    WMMA_F32(c0, a1, b10);
    WMMA_F32(c1, a1, b11);
    WMMA_F32(c2, a1, b12);
    WMMA_F32(c3, a1, b13);
    a1  = *(const v2f*)(Ap  + k + 12);
    b10 = *(const v2f*)(Bp0 + k + 12);
    b11 = *(const v2f*)(Bp1 + k + 12);
    b12 = *(const v2f*)(Bp2 + k + 12);
    b13 = *(const v2f*)(Bp3 + k + 12);
  }
  // tail: stage0 = data(Kd-8), stage1 = data(Kd-4)
  WMMA_F32(c0, a0, b00);
  WMMA_F32(c1, a0, b01);
  WMMA_F32(c2, a0, b02);
  WMMA_F32(c3, a0, b03);
  WMMA_F32(c0, a1, b10);
  WMMA_F32(c1, a1, b11);
  WMMA_F32(c2, a1, b12);
  WMMA_F32(c3, a1, b13);

  const float bv0 = bias[n0];
  const float bv1 = bias[n0 + 16];
  const float bv2 = bias[n0 + 32];
  const float bv3 = bias[n0 + 48];
#pragma unroll
  for (int i = 0; i < 8; ++i) {
    const int row = tileM + i + 8 * half;   // C layout: VGPR i -> M=i / i+8
    float v0 = c0[i] + bv0;
    float v1 = c1[i] + bv1;
    float v2 = c2[i] + bv2;
    float v3 = c3[i] + bv3;
    if (flags & 1) {
      const float* rr = R + (size_t)row * N;
      v0 += rr[n0];
      v1 += rr[n0 + 16];
      v2 += rr[n0 + 32];
      v3 += rr[n0 + 48];
    }
    if (flags & 2) {
      v0 = gelu_exact(v0);
      v1 = gelu_exact(v1);
      v2 = gelu_exact(v2);
      v3 = gelu_exact(v3);
    }
    size_t orow = (size_t)row;
    if (flags & 4) {
      const int bi = row / Tlen;
      orow = (size_t)bi * Kspk * Tlen + (size_t)kk * Tlen + (row % Tlen);
    }
    float* cr = C + orow * N;
    cr[n0]      = v0;
    cr[n0 + 16] = v1;
    cr[n0 + 32] = v2;
    cr[n0 + 48] = v3;
  }
}

// ---------------------------------------------------------------------------
// LayerNorm: Y[row,:] = (X-mu)/sqrt(var+eps)*g + b   (one block per row)
// ---------------------------------------------------------------------------
__global__ __launch_bounds__(256) void ln_kernel(
    const float* __restrict__ X, const float* __restrict__ g,
    const float* __restrict__ b, float* __restrict__ Y, int Dn)
{
  const int row = blockIdx.x;
  const int tid = threadIdx.x;
  const float* x = X + (size_t)row * Dn;
  float s = 0.f, s2 = 0.f;
  for (int d = tid; d < Dn; d += 256) { float v = x[d]; s += v; s2 += v * v; }
  __shared__ float sb0[256], sb1[256];
  sb0[tid] = s; sb1[tid] = s2; __syncthreads();
  for (int off = 128; off > 0; off >>= 1) {
    if (tid < off) { sb0[tid] += sb0[tid + off]; sb1[tid] += sb1[tid + off]; }
    __syncthreads();
  }
  const float mean = sb0[0] / Dn;
  const float var  = sb1[0] / Dn - mean * mean;
  const float inv  = rsqrtf(var + 1e-5f);
  float* y = Y + (size_t)row * Dn;
  for (int d = tid; d < Dn; d += 256) y[d] = (x[d] - mean) * inv * g[d] + b[d];
}

// ---------------------------------------------------------------------------
// Speaker mix weights from CTC logits:
//   A_k(b,t) = 1 - softmax_v(logits[k,b,t,:])[blank];  Wmix = softmax_k(A)
// ---------------------------------------------------------------------------
__global__ __launch_bounds__(256) void wmix_kernel(
    const float* __restrict__ logits, const int* __restrict__ blank_p,
    float* __restrict__ wmix, int BT, int V, int Kspk)
{
  const int row = blockIdx.x;
  const int tid = threadIdx.x;
  const int blank = blank_p[0];
  __shared__ float red[256];
  float a[8];
  for (int k2 = 0; k2 < Kspk; ++k2) {
    const float* L = logits + ((size_t)k2 * BT + row) * V;
    float m = -INFINITY;
    for (int v = tid; v < V; v += 256) m = fmaxf(m, L[v]);
    red[tid] = m; __syncthreads();
    for (int off = 128; off > 0; off >>= 1) {
      if (tid < off) red[tid] = fmaxf(red[tid], red[tid + off]);
      __syncthreads();
    }
    m = red[0]; __syncthreads();
    float s = 0.f;
    for (int v = tid; v < V; v += 256) s += expf(L[v] - m);
    red[tid] = s; __syncthreads();
    for (int off = 128; off > 0; off >>= 1) {
      if (tid < off) red[tid] += red[tid + off];
      __syncthreads();
    }
    s = red[0]; __syncthreads();
    a[k2] = 1.0f - expf(L[blank] - m) / s;
  }
  if (tid == 0) {
    float m2 = -INFINITY;
    for (int k2 = 0; k2 < Kspk; ++k2) m2 = fmaxf(m2, a[k2]);
    float ss = 0.f;
    float e[8];
    for (int k2 = 0; k2 < Kspk; ++k2) { e[k2] = expf(a[k2] - m2); ss += e[k2]; }
    for (int k2 = 0; k2 < Kspk; ++k2) wmix[(size_t)row * Kspk + k2] = e[k2] / ss;
  }
}

// ---------------------------------------------------------------------------
// Fused per-speaker stream prep: Xk = X*wmix[:,kk] + tag[kk]; Qin = LN(Xk)
// ---------------------------------------------------------------------------
__global__ __launch_bounds__(256) void stream_prep(
    const float* __restrict__ X, const float* __restrict__ wmix,
    const float* __restrict__ tags, const float* __restrict__ g,
    const float* __restrict__ bt, float* __restrict__ Xk,
    float* __restrict__ Qin, int Dn, int Kspk, int kk)
{
  const int row = blockIdx.x;
  const int tid = threadIdx.x;
  const float wv = wmix[(size_t)row * Kspk + kk];
  const float* x  = X    + (size_t)row * Dn;
  const float* tg = tags + (size_t)kk  * Dn;
  float s = 0.f, s2 = 0.f;
  for (int d = tid; d < Dn; d += 256) {
    float v = x[d] * wv + tg[d];
    Xk[(size_t)row * Dn + d] = v;
    s += v; s2 += v * v;
  }
  __shared__ float sb0[256], sb1[256];
  sb0[tid] = s; sb1[tid] = s2; __syncthreads();
  for (int off = 128; off > 0; off >>= 1) {
    if (tid < off) { sb0[tid] += sb0[tid + off]; sb1[tid] += sb1[tid + off]; }
    __syncthreads();
  }
  const float mean = sb0[0] / Dn;
  const float var  = sb1[0] / Dn - mean * mean;
  const float inv  = rsqrtf(var + 1e-5f);
  for (int d = tid; d < Dn; d += 256) {
    float v = x[d] * wv + tg[d];            // recompute, avoids reload hazard
    Qin[(size_t)row * Dn + d] = (v - mean) * inv * g[d] + bt[d];
  }
}

// ---------------------------------------------------------------------------
// Banded attention, hd=64, band width 2*BAND+1 (=49).
// One wave per (b,h,t): lane i owns score cols {i, i+32} and ctx dims
// {i, i+32}. q broadcast via LDS; dot products use b128 (float4) loads;
// v rows prefetched (global_prefetch) during the score pass.
// ---------------------------------------------------------------------------
__global__ __launch_bounds__(128) void banded_attn(
    const float* __restrict__ q, const float* __restrict__ kb,
    const float* __restrict__ vb, float* __restrict__ ctx,
    int T_, int Dn, int Hn, int band)
{
  const int hd   = 64;
  const int wid  = threadIdx.x >> 5;
  const int lane = threadIdx.x & 31;
  const int gw   = blockIdx.x * 4 + wid;
  const int t = gw % T_;
  const int h = (gw / T_) % Hn;
  const int b = gw / (T_ * Hn);

  __shared__ float lq[4][64];
  __shared__ float lat[4][64];

  const size_t rowbase = (size_t)(b * T_ + t) * Dn + (size_t)h * hd;
  lq[wid][lane]      = q[rowbase + lane];
  lq[wid][lane + 32] = q[rowbase + lane + 32];
  __syncthreads();

  const float scale = 0.125f;          // 1/sqrt(64)
  const int s0 = t - band;
  const int nb = 2 * band + 1;         // 49

  const float4* lq4 = (const float4*)&lq[wid][0];

  const int j0 = lane, j1 = lane + 32;
  const int sa = s0 + j0, sbi = s0 + j1;
  float sc0 = -INFINITY, sc1 = -INFINITY;
  if (j0 < nb && sa >= 0 && sa < T_) {
    const size_t kro = (size_t)(b * T_ + sa) * Dn + (size_t)h * hd;
    __builtin_prefetch(vb + kro, 0, 0);           // warm the v row for pass 2
    const float4* kp4 = (const float4*)(kb + kro);
    float acc = 0.f;
#pragma unroll
    for (int d4 = 0; d4 < 16; ++d4) {
      float4 kv = kp4[d4];
      float4 qv = lq4[d4];
      acc += qv.x * kv.x + qv.y * kv.y + qv.z * kv.z + qv.w * kv.w;
    }
    sc0 = acc * scale;
  }
  if (j1 < nb && sbi >= 0 && sbi < T_) {
    const size_t kro = (size_t)(b * T_ + sbi) * Dn + (size_t)h * hd;
    __builtin_prefetch(vb + kro, 0, 0);
    const float4* kp4 = (const float4*)(kb + kro);
    float acc = 0.f;
#pragma unroll
    for (int d4 = 0; d4 < 16; ++d4) {
      float4 kv = kp4[d4];
      float4 qv = lq4[d4];
      acc += qv.x * kv.x + qv.y * kv.y + qv.z * kv.z + qv.w * kv.w;
    }
    sc1 = acc * scale;
  }

  float m = fmaxf(sc0, sc1);
  for (int off = 16; off > 0; off >>= 1) m = fmaxf(m, __shfl_xor(m, off, 32));
  float e0 = expf(sc0 - m);            // -inf -> 0
  float e1 = expf(sc1 - m);
  float s = e0 + e1;
  for (int off = 16; off > 0; off >>= 1) s += __shfl_xor(s, off, 32);
  const float inv = 1.0f / s;
  lat[wid][j0] = e0 * inv;
  lat[wid][j1] = e1 * inv;
  __syncthreads();

  float a0 = 0.f, a1 = 0.f;
  const int d0 = lane, d1 = lane + 32;
  for (int j = 0; j < nb; ++j) {
    const int sidx = s0 + j;
    if (sidx < 0 || sidx >= T_) continue;
    const float wv = lat[wid][j];
    const float* vp = vb + (size_t)(b * T_ + sidx) * Dn + (size_t)h * hd;
    a0 += wv * vp[d0];
    a1 += wv * vp[d1];
  }
  float* cp = ctx + rowbase;
  cp[d0] = a0;
  cp[d1] = a1;
}

// ---------------------------------------------------------------------------
// Host-side orchestration
// ---------------------------------------------------------------------------
extern "C" void kernel_launch(void* const* d_in, const int* in_sizes, int n_in,
                              void* d_out, int out_size, void* d_ws, size_t ws_size,
                              hipStream_t stream) {
  (void)in_sizes; (void)n_in; (void)out_size; (void)ws_size;
  constexpr int B = 4, T = 1024, D_IN = 512, D = 512, K = 2, V = 500, H = 8, BAND = 24;
  constexpr int M = B * T;                 // 4096 token rows

  const float* x_m      = (const float*)d_in[0];
  const float* logits   = (const float*)d_in[1];
  const int*   blank_id = (const int*)  d_in[2];
  const float* proj_w   = (const float*)d_in[3];
  const float* proj_b   = (const float*)d_in[4];
  const float* spk_tags = (const float*)d_in[5];
  const float* lnq_g    = (const float*)d_in[6];
  const float* lnq_b    = (const float*)d_in[7];
  const float* lnkv_g   = (const float*)d_in[8];
  const float* lnkv_b   = (const float*)d_in[9];
  const float* in_w     = (const float*)d_in[10];
  const float* in_b     = (const float*)d_in[11];
  const float* out_w    = (const float*)d_in[12];
  const float* out_b    = (const float*)d_in[13];
  const float* ffn_ln_g = (const float*)d_in[14];
  const float* ffn_ln_b = (const float*)d_in[15];
  const float* ffn_w1   = (const float*)d_in[16];
  const float* ffn_b1   = (const float*)d_in[17];
  const float* ffn_w2   = (const float*)d_in[18];
  const float* ffn_b2   = (const float*)d_in[19];
  float* out = (float*)d_out;

  // workspace layout (floats)
  float* ws = (float*)d_ws;
  const size_t SZ = (size_t)M * D;         // 2,097,152 floats
  float* X    = ws;                        // mixed-feature projection
  float* kbuf = X    + SZ;
  float* vbuf = kbuf + SZ;
  float* tA   = vbuf + SZ;                 // KV / Xk
  float* tB   = tA   + SZ;                 // Qin / ctx / h
  float* tC   = tB   + SZ;                 // q / y
  float* t1   = tC   + SZ;                 // FFN hidden [M, 4D]
  float* wmix = t1   + 4 * SZ;             // [M, K]

  const dim3 blk(256);
  auto gg = [](int m, int n) { return dim3(m / 64, n / 128); };

  // X = x_m @ proj_w^T + proj_b
  gemm_wmma_f32<<<gg(M, D), blk, 0, stream>>>(x_m, proj_w, proj_b, nullptr, X,
                                              M, D, D_IN, 0, T, K, 0);
  // speaker mixing weights
  wmix_kernel<<<dim3(M), blk, 0, stream>>>(logits, blank_id, wmix, M, V, K);
  // KV = LN(X)
  ln_kernel<<<dim3(M), blk, 0, stream>>>(X, lnkv_g, lnkv_b, tA, D);
  // k, v projections (wk = in_w rows [D,2D), wv = rows [2D,3D))
  gemm_wmma_f32<<<gg(M, D), blk, 0, stream>>>(tA, in_w + (size_t)D * D, in_b + D,
                                              nullptr, kbuf, M, D, D, 0, T, K, 0);
  gemm_wmma_f32<<<gg(M, D), blk, 0, stream>>>(tA, in_w + (size_t)2 * D * D, in_b + 2 * D,
                                              nullptr, vbuf, M, D, D, 0, T, K, 0);

  for (int kk = 0; kk < K; ++kk) {
    // Xk (tA) and Qin (tB)
    stream_prep<<<dim3(M), blk, 0, stream>>>(X, wmix, spk_tags, lnq_g, lnq_b,
                                             tA, tB, D, K, kk);
    // q = Qin @ wq^T + bq  -> tC
    gemm_wmma_f32<<<gg(M, D), blk, 0, stream>>>(tB, in_w, in_b, nullptr, tC,
                                                M, D, D, 0, T, K, kk);
    // banded attention -> ctx (tB)
    banded_attn<<<dim3((B * H * T) / 4), dim3(128), 0, stream>>>(tC, kbuf, vbuf, tB,
                                                                 T, D, H, BAND);
    // y = Xk + ctx @ out_w^T + out_b -> tC
    gemm_wmma_f32<<<gg(M, D), blk, 0, stream>>>(tB, out_w, out_b, tA, tC,
                                                M, D, D, /*residual*/1, T, K, kk);
    // h = LN(y) -> tB
    ln_kernel<<<dim3(M), blk, 0, stream>>>(tC, ffn_ln_g, ffn_ln_b, tB, D);
    // t1 = gelu(h @ w1^T + b1)  [M, 4D]
    gemm_wmma_f32<<<gg(M, 4 * D), blk, 0, stream>>>(tB, ffn_w1, ffn_b1, nullptr, t1,
                                                    M, 4 * D, D, /*gelu*/2, T, K, kk);
    // out[b, kk*T + t, :] = y + t1 @ w2^T + b2
    gemm_wmma_f32<<<gg(M, D), blk, 0, stream>>>(t1, ffn_w2, ffn_b2, tC, out,
                                                M, D, 4 * D, /*res|concat*/1 | 4, T, K, kk);
  }
}